// SEDMetrics_31645319037286
// MI455X (gfx1250) — compile-verified
//
#include <hip/hip_runtime.h>

#define BN 256      // clips
#define CN 10       // classes
#define TN 10000    // frames (multiple of 4 -> clean float4 path)
#define EMAX 5000   // max contiguous events per track = ceil(T/2)
#define THREADS 256
#define CHUNK (4 * THREADS)   // 1024 elements per block iteration

typedef __attribute__((ext_vector_type(2))) float v2f;
typedef __attribute__((ext_vector_type(8))) float v8f;

// ---------------------------------------------------------------- init ws
__global__ void sed_init(int* ws) {
    int i = blockIdx.x * blockDim.x + threadIdx.x;
    if (i < 2 * BN) ws[i] = 0;
}

// ------------------------------------------------------- per-track kernel
// One 256-thread block (8 wave32s) per (b,c) track. Streaming O(T) pass with
// float4 (global_load_b128) loads: each thread owns 4 contiguous frames.
// Run-start detection -> shfl_up inclusive scan of per-thread start counts
// (+ 8-entry cross-wave scan) -> global event ids -> per-run-fragment merged
// LDS atomic accumulation of (pred_count | label_count<<16).
__global__ __launch_bounds__(THREADS) void sed_track(
        const float* __restrict__ preds, const float* __restrict__ gt,
        int* __restrict__ ws_tp, int* __restrict__ ws_n) {
    __shared__ int s_cnt[EMAX];
    __shared__ int s_ev[THREADS];    // last ev of each thread's 4 elements
    __shared__ int s_woff[8];
    __shared__ int s_base, s_carry;

    const int tid  = threadIdx.x;
    const int lane = tid & 31;
    const int wid  = tid >> 5;
    const int track = blockIdx.x;               // = b*CN + c  ([B,C,T] layout)
    const long off = (long)track * TN;          // 40000B/track -> 16B aligned
    const float4* __restrict__ p4p = (const float4*)(preds + off);
    const float4* __restrict__ g4p = (const float4*)(gt + off);

    for (int i = tid; i < EMAX; i += THREADS) s_cnt[i] = 0;
    if (tid == 0) { s_base = 0; s_carry = 0; }
    __syncthreads();

    for (int t0 = 0; t0 < TN; t0 += CHUNK) {
        const int t = t0 + tid * 4;
        float p[4] = {0.f, 0.f, 0.f, 0.f};
        float g[4] = {0.f, 0.f, 0.f, 0.f};
        if (t < TN) {                            // TN%4==0 -> whole vec valid
            float4 pv = p4p[t >> 2];
            float4 gv = g4p[t >> 2];
            p[0] = pv.x; p[1] = pv.y; p[2] = pv.z; p[3] = pv.w;
            g[0] = gv.x; g[1] = gv.y; g[2] = gv.z; g[3] = gv.w;
            if (t + CHUNK < TN) {                // gfx1250 global_prefetch_b8
                __builtin_prefetch(&preds[off + t + CHUNK], 0, 0);
                __builtin_prefetch(&gt[off + t + CHUNK], 0, 0);
            }
        }
        int ev[4];
#pragma unroll
        for (int j = 0; j < 4; ++j) ev[j] = (p[j] != 0.f) || (g[j] != 0.f);
        s_ev[tid] = ev[3];
        __syncthreads();

        const int prev = (tid == 0) ? s_carry : s_ev[tid - 1];
        int st[4];
        st[0] = ev[0] && !prev;
#pragma unroll
        for (int j = 1; j < 4; ++j) st[j] = ev[j] && !ev[j - 1];
        const int cnt = st[0] + st[1] + st[2] + st[3];

        int incl = cnt;                          // wave32 inclusive scan
#pragma unroll
        for (int o = 1; o < 32; o <<= 1) {
            int u = __shfl_up(incl, o);
            if (lane >= o) incl += u;
        }
        if (lane == 31) s_woff[wid] = incl;      // wave total
        __syncthreads();
        if (tid == 0) {   // exclusive scan of wave totals + carry update
            int acc = s_base;
            for (int w = 0; w < 8; ++w) { int cw = s_woff[w]; s_woff[w] = acc; acc += cw; }
            s_base  = acc;
            s_carry = s_ev[THREADS - 1];
        }
        __syncthreads();

        // per-run-fragment merged LDS atomics
        int c = s_woff[wid] + incl - cnt;        // starts strictly before me
        int cur = -1, acc = 0;
#pragma unroll
        for (int j = 0; j < 4; ++j) {
            c += st[j];
            if (ev[j]) {
                int eid = c - 1;                 // global event id in track
                if (eid != cur) {
                    if (cur >= 0) atomicAdd(&s_cnt[cur], acc);
                    cur = eid; acc = 0;
                }
                acc += (p[j] != 0.f ? 1 : 0) | (g[j] != 0.f ? (1 << 16) : 0);
            }
        }
        if (cur >= 0) atomicAdd(&s_cnt[cur], acc);
        __syncthreads();
    }

    // classify events
    const int nev = s_base;
    const float inv_thd = (float)(1.0 / 0.7);
    int tp = 0;
    for (int e = tid; e < nev; e += THREADS) {
        int v = s_cnt[e];
        float ps = (float)(v & 0xFFFF);
        float ls = (float)((unsigned)v >> 16);
        float r  = ps / (ls + 1e-7f);            // matches reference f32 math
        tp += (r >= 0.7f) && (r < inv_thd);
    }
    for (int o = 16; o; o >>= 1) tp += __shfl_xor(tp, o);
    if (lane == 0) s_ev[wid] = tp;               // reuse s_ev for reduction
    __syncthreads();
    if (tid == 0) {
        int s = 0;
        for (int w = 0; w < 8; ++w) s += s_ev[w];
        int b = track / CN;
        atomicAdd(&ws_tp[b], s);
        atomicAdd(&ws_n[b], nev);
    }
}

// --------------------------------------------------------- final reduction
// Single wave32 (EXEC all ones, as WMMA requires). Computes the 256 per-clip
// f-scores in f32, then sums them with 4 chained v_wmma_f32_16x16x4_f32:
// A = ones(16x4), B carries 64 f-scores each -> every row of D holds the
// column sums, so sum(c[0] over all 32 lanes) == 2 * total, exactly in f32,
// independent of the A/B VGPR layout details.
__global__ void sed_final(const int* __restrict__ ws_tp,
                          const int* __restrict__ ws_n,
                          float* __restrict__ out) {
    int lane = threadIdx.x;   // 0..31
    float f[8];
#pragma unroll
    for (int j = 0; j < 8; ++j) {
        int b = lane * 8 + j;
        float tpf = (float)ws_tp[b];
        float nf  = (float)ws_n[b];
        float den = 0.5f * tpf + 0.5f * nf;
        f[j] = den > 0.f ? tpf / den : 0.f;      // nan_to_num(0) semantics
    }
    v2f ones; ones[0] = 1.f; ones[1] = 1.f;
    v8f c = {};
#pragma unroll
    for (int i = 0; i < 4; ++i) {
        v2f bv; bv[0] = f[2 * i]; bv[1] = f[2 * i + 1];
        c = __builtin_amdgcn_wmma_f32_16x16x4_f32(
                /*neg_a=*/false, ones, /*neg_b=*/false, bv,
                /*c_mod=*/(short)0, c, /*reuse_a=*/false, /*reuse_b=*/false);
    }
    float v = c[0];
    for (int o = 16; o; o >>= 1) v += __shfl_xor(v, o);   // v == 2 * sum(f)
    if (lane == 0) out[0] = v * (0.5f / 256.f);           // exact /512
}

// ----------------------------------------------------------------- launch
extern "C" void kernel_launch(void* const* d_in, const int* in_sizes, int n_in,
                              void* d_out, int out_size, void* d_ws, size_t ws_size,
                              hipStream_t stream) {
    const float* preds = (const float*)d_in[0];
    const float* gt    = (const float*)d_in[1];
    int*   ws_tp = (int*)d_ws;        // [BN]
    int*   ws_n  = ws_tp + BN;        // [BN]
    float* out   = (float*)d_out;

    sed_init <<<2, 256, 0, stream>>>(ws_tp);
    sed_track<<<BN * CN, THREADS, 0, stream>>>(preds, gt, ws_tp, ws_n);
    sed_final<<<1, 32, 0, stream>>>(ws_tp, ws_n, out);
}